// NoauxTCRouter_2559800508620
// MI455X (gfx1250) — compile-verified
//
#include <hip/hip_runtime.h>
#include <hip/hip_bf16.h>
#include <math.h>

// ---------------------------------------------------------------------------
// NoauxTC MoE router for MI455X (gfx1250, wave32).
//   - TDM (tensor_load_to_lds) streams [8 tokens x 256 experts] f32 tiles
//     HBM -> LDS, double buffered, tracked with TENSORcnt.
//   - 1 wave per token; lane l owns experts l*8 .. l*8+7.
//   - Exact group top-2 / top-4-group / global top-8 with index tie-breaks
//     matching jax.lax.top_k (first index wins ties).
//   - Weights recovered as (corrected - bias[idx]) so the argmax butterflies
//     carry only (value, index): 2 ds-shuffles per level instead of 3.
// ---------------------------------------------------------------------------

#define N_EXPERTS      256
#define N_GROUP        8
#define TOPK_GROUP     4
#define TOP_K          8
#define EXP_PER_LANE   8
#define ROWS_PER_CHUNK 8              // 8 tokens per chunk = 1 per wave
#define BLOCK_THREADS  256            // 8 wave32s
#define CHUNK_FLOATS   (ROWS_PER_CHUNK * N_EXPERTS)
#define CHUNK_BYTES    (CHUNK_FLOATS * 4)

typedef __attribute__((ext_vector_type(4))) unsigned int v4u;
typedef __attribute__((ext_vector_type(8))) int          v8i;
typedef __attribute__((ext_vector_type(4))) int          v4i;

// Build a 2-D D# (see CDNA5 ISA) and issue the TDM DMA:
// tile = 256 x ROWS_PER_CHUNK f32, tensor_dim0_stride = 256 elements.
// amdgpu-toolchain (clang-23) 6-arg form: (v4u, v8i, v4i, v4i, v8i, i32 cpol).
__device__ __forceinline__ void tdm_load_chunk(const float* gptr,
                                               unsigned lds_byte_addr,
                                               unsigned tokens_total) {
  unsigned long long ga = (unsigned long long)(size_t)gptr;

  v4u g0;
  g0[0] = 1u;                                       // count=1 (valid), user mode
  g0[1] = lds_byte_addr;                            // lds_addr[31:0]
  g0[2] = (unsigned)(ga & 0xFFFFFFFFu);             // global_addr[31:0]
  g0[3] = (unsigned)((ga >> 32) & 0x01FFFFFFu)      // global_addr[56:32]
        | (2u << 30);                               // type = 2 ("image")

  v8i g1;
  g1[0] = (int)(2u << 16);                          // data_size=2 (4B), mask=0
  g1[1] = (int)(((unsigned)N_EXPERTS & 0xFFFFu) << 16);        // tensor_dim0 lo16
  g1[2] = (int)((((unsigned)N_EXPERTS >> 16) & 0xFFFFu)        // tensor_dim0 hi16
        | ((tokens_total & 0xFFFFu) << 16));                   // tensor_dim1 lo16
  g1[3] = (int)(((tokens_total >> 16) & 0xFFFFu)               // tensor_dim1 hi16
        | (((unsigned)N_EXPERTS & 0xFFFFu) << 16));            // tile_dim0 = 256
  g1[4] = (int)(ROWS_PER_CHUNK & 0xFFFF);           // tile_dim1 = 8, tile_dim2=0
  g1[5] = (int)N_EXPERTS;                           // tensor_dim0_stride[31:0]
  g1[6] = 0;                                        // stride hi, dim1_stride lo
  g1[7] = 0;

  v4i g2 = {0, 0, 0, 0};                            // 2-D tensor: unused dims
  v4i g3 = {0, 0, 0, 0};
  v8i g4 = {0, 0, 0, 0, 0, 0, 0, 0};                // extended words (unused, 2-D)

  __builtin_amdgcn_tensor_load_to_lds(g0, g1, g2, g3, g4, /*cpol=*/0);
}

__global__ __launch_bounds__(BLOCK_THREADS)
void noauxtc_router_kernel(const float* __restrict__ logits,
                           const float* __restrict__ bias,
                           float* __restrict__ out_w,   // [T, 8] weights
                           float* __restrict__ out_i,   // [T, 8] ids (as float)
                           int T, int nchunks) {
  __shared__ float smem[2 * CHUNK_FLOATS];            // 16 KB, double buffer

  const int tid   = threadIdx.x;
  const int lane  = tid & 31;
  const int wid   = tid >> 5;
  const int mygrp = lane >> 2;                        // 4 lanes per group

  // Per-lane expert bias (loaded once, held in 8 VGPRs).
  float bb[EXP_PER_LANE];
  {
    const float4* bp = (const float4*)(bias + lane * EXP_PER_LANE);
    float4 b0 = bp[0], b1 = bp[1];
    bb[0] = b0.x; bb[1] = b0.y; bb[2] = b0.z; bb[3] = b0.w;
    bb[4] = b1.x; bb[5] = b1.y; bb[6] = b1.z; bb[7] = b1.w;
  }

  const unsigned lds_base = (unsigned)(size_t)(&smem[0]);

  const int c0     = blockIdx.x;
  const int stride = gridDim.x;
  if (c0 >= nchunks) return;

  if (wid == 0)
    tdm_load_chunk(logits + (size_t)c0 * CHUNK_FLOATS, lds_base, (unsigned)T);

  int phase = 0;
  for (int c = c0; c < nchunks; c += stride, phase ^= 1) {
    const int nc = c + stride;
    if (wid == 0) {
      if (nc < nchunks)
        tdm_load_chunk(logits + (size_t)nc * CHUNK_FLOATS,
                       lds_base + (unsigned)((phase ^ 1) * CHUNK_BYTES),
                       (unsigned)T);
      if (nc < nchunks) __builtin_amdgcn_s_wait_tensorcnt(1);
      else              __builtin_amdgcn_s_wait_tensorcnt(0);
    }
    __syncthreads();                                  // chunk c resident in LDS

    const float* row = &smem[phase * CHUNK_FLOATS + wid * N_EXPERTS];
    float4 xa = ((const float4*)row)[lane * 2 + 0];
    float4 xb = ((const float4*)row)[lane * 2 + 1];
    float x[8] = {xa.x, xa.y, xa.z, xa.w, xb.x, xb.y, xb.z, xb.w};

    float cv[8];                                      // corrected scores
#pragma unroll
    for (int j = 0; j < 8; ++j)
      cv[j] = 1.0f / (1.0f + __expf(-x[j])) + bb[j];  // sigmoid + bias

    // ---- group score: sum of top-2 corrected scores within group ----
    float m1 = fmaxf(cv[0], cv[1]);
    float m2 = fminf(cv[0], cv[1]);
#pragma unroll
    for (int j = 2; j < 8; ++j) {
      float v   = cv[j];
      float nm1 = fmaxf(m1, v);
      m2 = fmaxf(m2, fminf(m1, v));
      m1 = nm1;
    }
#pragma unroll
    for (int msk = 1; msk <= 2; msk <<= 1) {          // merge across 4 lanes
      float o1 = __shfl_xor(m1, msk, 32);
      float o2 = __shfl_xor(m2, msk, 32);
      float hi = fmaxf(m1, o1);
      float lo = fminf(m1, o1);
      m1 = hi;
      m2 = fmaxf(lo, fmaxf(m2, o2));
    }
    const float gs = m1 + m2;

    // ---- top-4 groups: rank my group among the 8 (ties -> lower index) ----
    int rank = 0;
#pragma unroll
    for (int g = 0; g < N_GROUP; ++g) {
      float og = __shfl(gs, g * 4, 32);
      rank += ((og > gs) || (og == gs && g < mygrp)) ? 1 : 0;
    }
    const bool selected = (rank < TOPK_GROUP);

    float cm[8];
#pragma unroll
    for (int j = 0; j < 8; ++j) cm[j] = selected ? cv[j] : 0.0f;

    // ---- exact global top-8 (value, then lowest index on ties) ----
    float bvk[TOP_K];
    int   widx[TOP_K];
#pragma unroll
    for (int k = 0; k < TOP_K; ++k) {
      float bv = cm[0];
      int   bidx = lane * 8;
#pragma unroll
      for (int j = 1; j < 8; ++j) {
        bool t = cm[j] > bv;
        bv   = t ? cm[j] : bv;
        bidx = t ? lane * 8 + j : bidx;
      }
#pragma unroll
      for (int msk = 1; msk < 32; msk <<= 1) {        // wave argmax butterfly
        float ov = __shfl_xor(bv, msk, 32);
        int   oi = __shfl_xor(bidx, msk, 32);
        bool t = (ov > bv) || (ov == bv && oi < bidx);
        bv   = t ? ov : bv;
        bidx = t ? oi : bidx;
      }
      bvk[k]  = bv;                                   // winner corrected score
      widx[k] = bidx;                                 // winner expert id
#pragma unroll
      for (int j = 0; j < 8; ++j)                     // remove winner
        cm[j] = (bidx == lane * 8 + j) ? -INFINITY : cm[j];
    }

    // ---- recover uncorrected sigmoid weights: s = corrected - bias[idx] ----
    // 8 wave-uniform bias gathers, issued back-to-back (latency overlapped,
    // coalesces to a single request each, WGP$/L2 resident).
    float sv[TOP_K];
#pragma unroll
    for (int k = 0; k < TOP_K; ++k) sv[k] = bvk[k] - bias[widx[k]];

    // ---- renormalize * 2.5 and store ----
    float sum = 0.0f;
#pragma unroll
    for (int k = 0; k < TOP_K; ++k) sum += sv[k];
    const float scale = 2.5f / (sum + 1e-20f);

    float myw = 0.0f;
    int   myi = 0;
#pragma unroll
    for (int k = 0; k < TOP_K; ++k) {
      bool t = (lane == k);
      myw = t ? sv[k] * scale : myw;
      myi = t ? widx[k] : myi;
    }
    const int token = c * ROWS_PER_CHUNK + wid;
    if (lane < TOP_K && token < T) {
      out_w[(size_t)token * TOP_K + lane] = myw;
      out_i[(size_t)token * TOP_K + lane] = (float)myi;
    }
    __syncthreads();                                  // buffer reuse fence
  }
}

extern "C" void kernel_launch(void* const* d_in, const int* in_sizes, int n_in,
                              void* d_out, int out_size, void* d_ws, size_t ws_size,
                              hipStream_t stream) {
  const float* logits = (const float*)d_in[0];   // [T, 256] f32
  const float* bias   = (const float*)d_in[1];   // [256]    f32
  const int T = in_sizes[0] / N_EXPERTS;
  const int nchunks = (T + ROWS_PER_CHUNK - 1) / ROWS_PER_CHUNK;

  float* out_w = (float*)d_out;                  // [T, 8] weights
  float* out_i = out_w + (size_t)T * TOP_K;      // [T, 8] ids as float

  int grid = nchunks < 4096 ? nchunks : 4096;    // grid-stride over chunks
  noauxtc_router_kernel<<<grid, BLOCK_THREADS, 0, stream>>>(
      logits, bias, out_w, out_i, T, nchunks);
}